// Attention_10247791968780
// MI455X (gfx1250) — compile-verified
//
#include <hip/hip_runtime.h>
#include <stdint.h>

#define NUM_HEADS       32
#define NUM_KV_HEADS    8
#define HEAD_DIM        128
#define BLOCK_SIZE      16
#define BLOCKS_PER_SEQ  256
#define BATCH           16
#define SPLITS          16
#define BLOCKS_PER_SPLIT 16      // 256 / SPLITS
#define N_REP           4
#define ATT_SCALE       0.08838834764831845f

#define KPAD 132                 // padded row stride (floats) for Q/K/V tiles -> conflict-free
#define PPAD 18                  // padded row stride for P tile

typedef float v2f __attribute__((ext_vector_type(2)));
typedef float v8f __attribute__((ext_vector_type(8)));

__device__ __forceinline__ unsigned lds_off(const void* p) {
  // flat LDS pointer: addr[31:0] is the LDS byte offset (ISA 10.2 aperture rules)
  return (unsigned)(uintptr_t)p;
}

__global__ __launch_bounds__(32)
void attn_split_kernel(const float* __restrict__ q,
                       const float* __restrict__ knew,
                       const float* __restrict__ vnew,
                       const float* __restrict__ k_cache,
                       const float* __restrict__ v_cache,
                       const int*   __restrict__ block_tables,
                       const int*   __restrict__ context_lens,
                       const int*   __restrict__ slot_mapping,
                       float*       __restrict__ part_o,
                       float*       __restrict__ part_ml) {
  __shared__ __align__(16) float Qs[16 * KPAD];
  __shared__ __align__(16) float Ks[2][16 * KPAD];
  __shared__ __align__(16) float Vs[2][16 * KPAD];
  __shared__ __align__(16) float Ps[16 * PPAD];

  const int split = blockIdx.x;
  const int kvh   = blockIdx.y;
  const int b     = blockIdx.z;
  const int lane  = (int)(threadIdx.x & 31);
  const int l16   = lane & 15;
  const int hf    = lane >> 4;

  // ---- stage Q into LDS: rows 0..3 = real heads of this kv group, rows 4..15 = 0 ----
  #pragma unroll
  for (int e = 0; e < 64; ++e) {
    int idx = lane + 32 * e;           // 0..2047
    int r = idx >> 7, d = idx & 127;
    Qs[r * KPAD + d] = (r < N_REP) ? q[((b * NUM_HEADS + kvh * N_REP + r) << 7) + d] : 0.0f;
  }
  __builtin_amdgcn_wave_barrier();

  const int ctx = context_lens[b];
  const int ctx_blocks = (ctx + BLOCK_SIZE - 1) >> 4;
  int nb = ctx_blocks - split * BLOCKS_PER_SPLIT;
  nb = nb < 0 ? 0 : (nb > BLOCKS_PER_SPLIT ? BLOCKS_PER_SPLIT : nb);

  // preload slot mapping (scalar, compile-time indexed)
  int sm_r[BATCH];
  #pragma unroll
  for (int b2 = 0; b2 < BATCH; ++b2) sm_r[b2] = slot_mapping[b2];

  v8f acc[8] = {};
  float m_st[8], l_st[8];
  #pragma unroll
  for (int j = 0; j < 8; ++j) { m_st[j] = -1e30f; l_st[j] = 0.0f; }

  // async DMA of one 16x128 f32 tile (8KB) into LDS: 16 x b128-per-lane, row by row
  auto issue_tile = [&](const float* gbase, unsigned lbase) {
    uint64_t ga = (uint64_t)(uintptr_t)gbase + (unsigned)(lane * 16);
    unsigned  la = lbase + (unsigned)(lane * 16);
    #pragma unroll
    for (int j = 0; j < 16; ++j) {
      asm volatile("global_load_async_to_lds_b128 %0, %1, off"
                   :: "v"(la), "v"(ga) : "memory");
      la += KPAD * 4;                          // 528 B padded LDS row
      ga += NUM_KV_HEADS * HEAD_DIM * 4;       // 4096 B token stride in cache
    }
  };

  const int btbase = b * BLOCKS_PER_SEQ + split * BLOCKS_PER_SPLIT;
  int buf = 0;
  int cur_phys = 0;
  if (nb > 0) {
    cur_phys = block_tables[btbase];
    const size_t cbase = (((size_t)cur_phys * BLOCK_SIZE) * NUM_KV_HEADS + kvh) * HEAD_DIM;
    issue_tile(k_cache + cbase, lds_off(&Ks[0][0]));
    issue_tile(v_cache + cbase, lds_off(&Vs[0][0]));
  }

  for (int i = 0; i < nb; ++i) {
    asm volatile("s_wait_asynccnt 0x0" ::: "memory");   // tile for block i landed in LDS

    // ---- patch dirty slots (reference scatters new k/v into cache before gather) ----
    #pragma unroll
    for (int b2 = 0; b2 < BATCH; ++b2) {
      if ((sm_r[b2] >> 4) == cur_phys) {                // wave-uniform branch, rare
        int pos = sm_r[b2] & 15;
        float4 kf = *(const float4*)&knew[((b2 * NUM_KV_HEADS + kvh) << 7) + lane * 4];
        float4 vf = *(const float4*)&vnew[((b2 * NUM_KV_HEADS + kvh) << 7) + lane * 4];
        *(float4*)&Ks[buf][pos * KPAD + lane * 4] = kf;
        *(float4*)&Vs[buf][pos * KPAD + lane * 4] = vf;
      }
    }
    __builtin_amdgcn_wave_barrier();

    // ---- prefetch next block into other buffer (hidden behind compute) ----
    int nxt_phys = 0;
    if (i + 1 < nb) {
      nxt_phys = block_tables[btbase + i + 1];
      const size_t nbase = (((size_t)nxt_phys * BLOCK_SIZE) * NUM_KV_HEADS + kvh) * HEAD_DIM;
      issue_tile(k_cache + nbase, lds_off(&Ks[buf ^ 1][0]));
      issue_tile(v_cache + nbase, lds_off(&Vs[buf ^ 1][0]));
    }

    // ---- QK^T: scores[head(16), token(16)] via 32 x V_WMMA_F32_16X16X4_F32 ----
    const float* Kt = &Ks[buf][0];
    v8f s = {};
    #pragma unroll
    for (int c = 0; c < 32; ++c) {
      const int d0 = 4 * c + 2 * hf;
      v2f a;  a.x  = Qs[l16 * KPAD + d0];  a.y  = Qs[l16 * KPAD + d0 + 1];
      v2f bm; bm.x = Kt[l16 * KPAD + d0];  bm.y = Kt[l16 * KPAD + d0 + 1];
      s = __builtin_amdgcn_wmma_f32_16x16x4_f32(false, a, false, bm, (short)0, s, false, false);
    }

    // ---- mask + online softmax (rows = heads across VGPRs, cols = tokens across lanes) ----
    const int  tok0  = (split * BLOCKS_PER_SPLIT + i) * BLOCK_SIZE;
    const bool valid = (tok0 + l16) < ctx;
    v8f sc_row;
    #pragma unroll
    for (int j = 0; j < 8; ++j) {
      float sv = valid ? s[j] * ATT_SCALE : -1e30f;
      float r = sv;
      r = fmaxf(r, __shfl_xor(r, 1, 32));
      r = fmaxf(r, __shfl_xor(r, 2, 32));
      r = fmaxf(r, __shfl_xor(r, 4, 32));
      r = fmaxf(r, __shfl_xor(r, 8, 32));
      float mn  = fmaxf(m_st[j], r);
      float scf = __expf(m_st[j] - mn);
      float p   = __expf(sv - mn);
      float rs  = p;
      rs += __shfl_xor(rs, 1, 32);
      rs += __shfl_xor(rs, 2, 32);
      rs += __shfl_xor(rs, 4, 32);
      rs += __shfl_xor(rs, 8, 32);
      l_st[j]   = l_st[j] * scf + rs;
      m_st[j]   = mn;
      sc_row[j] = scf;
      Ps[(j + 8 * hf) * PPAD + l16] = p;   // D-layout -> row-major P tile in LDS
    }
    __builtin_amdgcn_wave_barrier();

    #pragma unroll
    for (int n = 0; n < 8; ++n) acc[n] *= sc_row;

    // ---- PV: out[head(16), d(128)] via 32 x V_WMMA_F32_16X16X4_F32 ----
    const float* Vt = &Vs[buf][0];
    #pragma unroll
    for (int tc = 0; tc < 4; ++tc) {
      const int t0 = 4 * tc + 2 * hf;
      v2f ap; ap.x = Ps[l16 * PPAD + t0]; ap.y = Ps[l16 * PPAD + t0 + 1];
      #pragma unroll
      for (int n = 0; n < 8; ++n) {
        v2f bv;
        bv.x = Vt[t0 * KPAD + n * 16 + l16];
        bv.y = Vt[(t0 + 1) * KPAD + n * 16 + l16];
        acc[n] = __builtin_amdgcn_wmma_f32_16x16x4_f32(false, ap, false, bv, (short)0, acc[n], false, false);
      }
    }

    cur_phys = nxt_phys;
    buf ^= 1;
  }

  // ---- write per-split partials (unnormalized acc + m/l per real head) ----
  const int pidx = (b * NUM_KV_HEADS + kvh) * SPLITS + split;
  float* op = part_o + (size_t)pidx * (N_REP * HEAD_DIM);
  if (hf == 0) {
    #pragma unroll
    for (int j = 0; j < N_REP; ++j)
      #pragma unroll
      for (int n = 0; n < 8; ++n)
        op[j * HEAD_DIM + n * 16 + l16] = acc[n][j];
  }
  if (lane == 0) {
    float* mp = part_ml + (size_t)pidx * (2 * N_REP);
    #pragma unroll
    for (int j = 0; j < N_REP; ++j) { mp[2 * j] = m_st[j]; mp[2 * j + 1] = l_st[j]; }
  }
}

__global__ __launch_bounds__(128)
void combine_kernel(const float* __restrict__ part_o,
                    const float* __restrict__ part_ml,
                    float* __restrict__ out) {
  const int bk  = blockIdx.x;            // b * 8 + kvh
  const int b   = bk >> 3, kvh = bk & 7;
  const int tid = (int)threadIdx.x;
  #pragma unroll
  for (int e = 0; e < 4; ++e) {
    int idx = tid + 128 * e;             // 0..511 -> (head, d)
    int h = idx >> 7, d = idx & 127;
    float M = -1e30f;
    #pragma unroll
    for (int s2 = 0; s2 < SPLITS; ++s2)
      M = fmaxf(M, part_ml[(size_t)(bk * SPLITS + s2) * (2 * N_REP) + 2 * h]);
    float L = 0.0f, o = 0.0f;
    #pragma unroll
    for (int s2 = 0; s2 < SPLITS; ++s2) {
      size_t mb = (size_t)(bk * SPLITS + s2) * (2 * N_REP) + 2 * h;
      float w = __expf(part_ml[mb] - M);
      L += w * part_ml[mb + 1];
      o += w * part_o[((size_t)(bk * SPLITS + s2) * N_REP + h) * HEAD_DIM + d];
    }
    out[((b * NUM_HEADS + kvh * N_REP + h) << 7) + d] = o / L;
  }
}

extern "C" void kernel_launch(void* const* d_in, const int* in_sizes, int n_in,
                              void* d_out, int out_size, void* d_ws, size_t ws_size,
                              hipStream_t stream) {
  const float* q  = (const float*)d_in[0];
  const float* k  = (const float*)d_in[1];
  const float* v  = (const float*)d_in[2];
  const float* kc = (const float*)d_in[3];
  const float* vc = (const float*)d_in[4];
  const int* bt   = (const int*)d_in[5];
  const int* cl   = (const int*)d_in[6];
  const int* sm   = (const int*)d_in[7];
  float* out = (float*)d_out;

  float* ws      = (float*)d_ws;
  float* part_o  = ws;                                                       // 2048 * 512 floats = 4 MB
  float* part_ml = ws + (size_t)BATCH * NUM_KV_HEADS * SPLITS * N_REP * HEAD_DIM; // + 2048*8 floats

  dim3 g1(SPLITS, NUM_KV_HEADS, BATCH);
  attn_split_kernel<<<g1, 32, 0, stream>>>(q, k, v, kc, vc, bt, cl, sm, part_o, part_ml);
  combine_kernel<<<BATCH * NUM_KV_HEADS, 128, 0, stream>>>(part_o, part_ml, out);
}